// MultiHeadAttentionWithEdgeBias2D_17721035063345
// MI455X (gfx1250) — compile-verified
//
#include <hip/hip_runtime.h>
#include <math.h>

#define HIDDEN 128
#define HEADS 8
#define HEAD_DIM 16

typedef __attribute__((ext_vector_type(2))) float v2f;
typedef __attribute__((ext_vector_type(8))) float v8f;

// ---------- helpers ----------
__device__ __forceinline__ unsigned fkey(float f) {
    unsigned u = __float_as_uint(f);
    return (u & 0x80000000u) ? ~u : (u | 0x80000000u);   // order-preserving map
}
__device__ __forceinline__ float fdecode(unsigned kk) {
    unsigned u = (kk & 0x80000000u) ? (kk & 0x7fffffffu) : ~kk;
    return __uint_as_float(u);
}

// ---------- pass 1: q/k/v node projections (fp32 WMMA 16x16x4) ----------
// one wave = one (16-node tile, matrix in {q,k,v}, 16-col tile)
__global__ __launch_bounds__(256) void qkv_proj_kernel(
    const float* __restrict__ x,
    const float* __restrict__ qw, const float* __restrict__ qb,
    const float* __restrict__ kw, const float* __restrict__ kb,
    const float* __restrict__ vw, const float* __restrict__ vb,
    float* __restrict__ q, float* __restrict__ k, float* __restrict__ v,
    int n_nodes, int n_tiles)
{
    const int lane = threadIdx.x & 31;
    const int job  = blockIdx.x * (blockDim.x >> 5) + (threadIdx.x >> 5);
    if (job >= n_tiles * 24) return;                  // wave-uniform exit

    const int tile = job / 24;
    const int sub  = job % 24;
    const int mat  = sub >> 3;                        // 0=q 1=k 2=v
    const int col0 = (sub & 7) << 4;

    const float* W; const float* B; float* O;
    if (mat == 0)      { W = qw; B = qb; O = q; }
    else if (mat == 1) { W = kw; B = kb; O = k; }
    else               { W = vw; B = vb; O = v; }

    const int m   = lane & 15;
    const int hi  = lane >> 4;
    const int khi = hi << 1;                          // 0 or 2

    int nr = tile * 16 + m; if (nr >= n_nodes) nr = n_nodes - 1;
    const float* arow = x + (size_t)nr * HIDDEN;                   // A: node rows
    const float* brow = W + (size_t)(col0 + m) * HIDDEN;           // B: W^T -> W rows

    v8f acc = {};
#pragma unroll
    for (int k0 = 0; k0 < HIDDEN; k0 += 4) {
        v2f a = *(const v2f*)(arow + k0 + khi);       // A[m][k0+khi .. +1]
        v2f b = *(const v2f*)(brow + k0 + khi);       // B[k][col0+m] = W[col0+m][k]
        acc = __builtin_amdgcn_wmma_f32_16x16x4_f32(false, a, false, b,
                                                    (short)0, acc, false, false);
    }

    const float bias = B[col0 + m];
    const int mrow0 = tile * 16 + hi * 8;             // D: M = r + 8*hi, N = m
#pragma unroll
    for (int r = 0; r < 8; ++r) {
        int row = mrow0 + r;
        if (row < n_nodes)
            O[(size_t)row * HIDDEN + col0 + m] = acc[r] + bias;
    }
}

// ---------- pass 2: edge bias projection bias[E,8] = edge_emb @ eb_w.T (+eb_b) ----------
// one wave = 16 edges; fp32 WMMA with B columns 8..15 zeroed.
// edge_emb is an 819MB one-shot stream -> non-temporal loads keep it out of L2.
__global__ __launch_bounds__(256) void edge_bias_kernel(
    const float* __restrict__ ee, const float* __restrict__ ebw,
    const float* __restrict__ ebb, float* __restrict__ bias_out,
    int n_edges, int e_tiles)
{
    const int lane = threadIdx.x & 31;
    const int job  = blockIdx.x * (blockDim.x >> 5) + (threadIdx.x >> 5);
    if (job >= e_tiles) return;                       // wave-uniform exit

    const int m   = lane & 15;
    const int hi  = lane >> 4;
    const int khi = hi << 1;

    int er = job * 16 + m; if (er >= n_edges) er = n_edges - 1;
    const float* arow = ee + (size_t)er * HIDDEN;                  // A: edge rows
    const float* brow = ebw + (size_t)(m & 7) * HIDDEN;            // B: eb_w rows
    const float bmask = (m < 8) ? 1.0f : 0.0f;        // zero out cols 8..15

    v8f acc = {};
#pragma unroll
    for (int k0 = 0; k0 < HIDDEN; k0 += 4) {
        const int kk = k0 + khi;
        v2f a = __builtin_nontemporal_load((const v2f*)(arow + kk));
        v2f b = *(const v2f*)(brow + kk);
        b.x *= bmask; b.y *= bmask;
        acc = __builtin_amdgcn_wmma_f32_16x16x4_f32(false, a, false, b,
                                                    (short)0, acc, false, false);
    }

    if (m < 8) {
        const float bb = ebb[m];
        const int e0 = job * 16 + hi * 8;
#pragma unroll
        for (int r = 0; r < 8; ++r) {
            int e = e0 + r;
            if (e < n_edges)
                bias_out[(size_t)e * HEADS + m] = acc[r] + bb;
        }
    }
}

// ---------- pass 3: per-edge q.k scores + segment max ----------
// one wave = 4 edges; lane = (edge = lane>>3, head = lane&7); no shuffles needed
__global__ __launch_bounds__(256) void edge_scores_kernel(
    const int* __restrict__ ei,
    const float* __restrict__ q, const float* __restrict__ k,
    const float* __restrict__ bias, float* __restrict__ scores,
    unsigned* __restrict__ segmax, int n_edges)
{
    const int lane = threadIdx.x & 31;
    const int wq = blockIdx.x * (blockDim.x >> 5) + (threadIdx.x >> 5);
    const int h = lane & 7;
    int e = wq * 4 + (lane >> 3);
    if (e >= n_edges) e = n_edges - 1;                // clamp: duplicate work is benign

    const int row = ei[e];
    const int col = ei[n_edges + e];
    const float* qp = q + (size_t)row * HIDDEN + h * HEAD_DIM;
    const float* kp = k + (size_t)col * HIDDEN + h * HEAD_DIM;

    float s = 0.0f;
#pragma unroll
    for (int t = 0; t < HEAD_DIM; t += 4) {
        float4 a = *(const float4*)(qp + t);
        float4 b = *(const float4*)(kp + t);
        s += a.x * b.x + a.y * b.y + a.z * b.z + a.w * b.w;
    }
    s = s * 0.25f + bias[(size_t)e * HEADS + h];      // 1/sqrt(16) = 0.25

    scores[(size_t)e * HEADS + h] = s;                // coalesced 128B wave store
    atomicMax(&segmax[(size_t)row * HEADS + h], fkey(s));
}

// ---------- pass 4: exp weights, segment sum, unnormalized scatter of v ----------
// one wave per edge; lane covers dims [lane*4, lane*4+4), head = lane>>2
__global__ __launch_bounds__(256) void edge_scatter_kernel(
    const int* __restrict__ ei, const float* __restrict__ v,
    const float* __restrict__ scores, const unsigned* __restrict__ segmax,
    float* __restrict__ segsum, float* __restrict__ agg, int n_edges)
{
    const int lane = threadIdx.x & 31;
    const int e = blockIdx.x * (blockDim.x >> 5) + (threadIdx.x >> 5);
    if (e >= n_edges) return;

    const int row = ei[e];
    const int col = ei[n_edges + e];
    const int h = lane >> 2;

    const float s = scores[(size_t)e * HEADS + h];
    const float m = fdecode(segmax[(size_t)row * HEADS + h]);
    const float w = __expf(s - m);

    if ((lane & 3) == 0)
        __hip_atomic_fetch_add(&segsum[(size_t)row * HEADS + h], w,
                               __ATOMIC_RELAXED, __HIP_MEMORY_SCOPE_AGENT);

    const float4 vv = *(const float4*)(v + (size_t)col * HIDDEN + lane * 4);
    float* dst = agg + (size_t)row * HIDDEN + lane * 4;
    __hip_atomic_fetch_add(dst + 0, w * vv.x, __ATOMIC_RELAXED, __HIP_MEMORY_SCOPE_AGENT);
    __hip_atomic_fetch_add(dst + 1, w * vv.y, __ATOMIC_RELAXED, __HIP_MEMORY_SCOPE_AGENT);
    __hip_atomic_fetch_add(dst + 2, w * vv.z, __ATOMIC_RELAXED, __HIP_MEMORY_SCOPE_AGENT);
    __hip_atomic_fetch_add(dst + 3, w * vv.w, __ATOMIC_RELAXED, __HIP_MEMORY_SCOPE_AGENT);
}

// ---------- pass 5: normalize + output projection (fp32 WMMA 16x16x4) ----------
__global__ __launch_bounds__(256) void out_proj_kernel(
    const float* __restrict__ agg, const float* __restrict__ segsum,
    const float* __restrict__ ow, const float* __restrict__ ob,
    float* __restrict__ out, int n_nodes, int n_tiles)
{
    const int lane = threadIdx.x & 31;
    const int job  = blockIdx.x * (blockDim.x >> 5) + (threadIdx.x >> 5);
    if (job >= n_tiles * 8) return;

    const int tile = job >> 3;
    const int col0 = (job & 7) << 4;

    const int m   = lane & 15;
    const int hi  = lane >> 4;
    const int khi = hi << 1;

    int nr = tile * 16 + m; if (nr >= n_nodes) nr = n_nodes - 1;
    const float* arow = agg + (size_t)nr * HIDDEN;
    const float* srow = segsum + (size_t)nr * HEADS;
    const float* brow = ow + (size_t)(col0 + m) * HIDDEN;

    float inv[HEADS];
#pragma unroll
    for (int hh = 0; hh < HEADS; ++hh) inv[hh] = 1.0f / (srow[hh] + 1e-10f);

    v8f acc = {};
#pragma unroll
    for (int k0 = 0; k0 < HIDDEN; k0 += 4) {
        const int kk = k0 + khi;                      // kk and kk+1 share a head
        v2f a = *(const v2f*)(arow + kk);
        const float sc = inv[kk >> 4];
        a.x *= sc; a.y *= sc;
        v2f b = *(const v2f*)(brow + kk);
        acc = __builtin_amdgcn_wmma_f32_16x16x4_f32(false, a, false, b,
                                                    (short)0, acc, false, false);
    }

    const float bias = ob[col0 + m];
    const int mrow0 = tile * 16 + hi * 8;
#pragma unroll
    for (int r = 0; r < 8; ++r) {
        int row = mrow0 + r;
        if (row < n_nodes)
            out[(size_t)row * HIDDEN + col0 + m] = acc[r] + bias;
    }
}

extern "C" void kernel_launch(void* const* d_in, const int* in_sizes, int n_in,
                              void* d_out, int out_size, void* d_ws, size_t ws_size,
                              hipStream_t stream) {
    const float* x   = (const float*)d_in[0];
    const int*   ei  = (const int*)  d_in[1];
    const float* ee  = (const float*)d_in[2];
    const float* qw  = (const float*)d_in[3];
    const float* qb  = (const float*)d_in[4];
    const float* kw  = (const float*)d_in[5];
    const float* kb  = (const float*)d_in[6];
    const float* vw  = (const float*)d_in[7];
    const float* vb  = (const float*)d_in[8];
    const float* ow  = (const float*)d_in[9];
    const float* ob  = (const float*)d_in[10];
    const float* ebw = (const float*)d_in[11];
    const float* ebb = (const float*)d_in[12];

    const int n_nodes = in_sizes[0] / HIDDEN;
    const int n_edges = in_sizes[1] / 2;
    const int n_tiles = (n_nodes + 15) / 16;
    const int e_tiles = (n_edges + 15) / 16;

    // workspace layout: q | k | v | scores | bias | segmax | segsum | agg
    float*    q      = (float*)d_ws;
    float*    kbuf   = q    + (size_t)n_nodes * HIDDEN;
    float*    v      = kbuf + (size_t)n_nodes * HIDDEN;
    float*    scores = v    + (size_t)n_nodes * HIDDEN;
    float*    bias   = scores + (size_t)n_edges * HEADS;
    unsigned* segmax = (unsigned*)(bias + (size_t)n_edges * HEADS);
    float*    segsum = (float*)(segmax + (size_t)n_nodes * HEADS);
    float*    agg    = segsum + (size_t)n_nodes * HEADS;

    // zero segmax (ordered-key 0 == minimum) + segsum + agg in one contiguous memset
    hipMemsetAsync(segmax, 0,
                   ((size_t)n_nodes * HEADS * 2 + (size_t)n_nodes * HIDDEN) * sizeof(float),
                   stream);

    {   // pass 1: QKV projections
        const int jobs = n_tiles * 24;
        dim3 g((jobs + 7) / 8), b(256);
        qkv_proj_kernel<<<g, b, 0, stream>>>(x, qw, qb, kw, kb, vw, vb,
                                             q, kbuf, v, n_nodes, n_tiles);
    }
    {   // pass 2: edge bias projection (WMMA, non-temporal edge_emb stream)
        dim3 g((e_tiles + 7) / 8), b(256);
        edge_bias_kernel<<<g, b, 0, stream>>>(ee, ebw, ebb, bias, n_edges, e_tiles);
    }
    {   // pass 3: q.k scores + segment max (4 edges per wave)
        const int waves = (n_edges + 3) / 4;
        dim3 g((waves + 7) / 8), b(256);
        edge_scores_kernel<<<g, b, 0, stream>>>(ei, q, kbuf, bias,
                                                scores, segmax, n_edges);
    }
    {   // pass 4: exp / segment-sum / weighted scatter of v
        dim3 g((n_edges + 7) / 8), b(256);
        edge_scatter_kernel<<<g, b, 0, stream>>>(ei, v, scores, segmax,
                                                 segsum, agg, n_edges);
    }
    {   // pass 5: normalize + output projection
        const int jobs = n_tiles * 8;
        dim3 g((jobs + 7) / 8), b(256);
        out_proj_kernel<<<g, b, 0, stream>>>(agg, segsum, ow, ob,
                                             (float*)d_out, n_nodes, n_tiles);
    }
    (void)n_in; (void)out_size; (void)ws_size;
}